// UpSample_53884659695889
// MI455X (gfx1250) — compile-verified
//
#include <hip/hip_runtime.h>
#include <math.h>

typedef float v2f __attribute__((ext_vector_type(2)));
typedef float v8f __attribute__((ext_vector_type(8)));

// Problem constants (from setup_inputs: scale_x10=15 -> scale=1.5)
#define Nb    8
#define INC   64
#define OUTC  3
#define Hin   96
#define Win   96
#define HWs   (Hin*Win)      // 9216
#define OUTH  144
#define OUTW  144
#define CYC   3
#define KTOT  576            // 9*INC
#define NPAD  16
#define PHB   (KTOT*NPAD)    // 9216 floats per phase B tile
#define B_SZ  (9*PHB)        // 82944 floats
#define RS_OFF B_SZ          // int table [OUTH][3]
#define VH_OFF (B_SZ + OUTH*3) // float table [OUTH][3]

// ---- _proj helpers (scale=1.5, scale_int=2, in_len=96, out_len=144) ----
__device__ __forceinline__ long coord_of(int o) {
  double c = (double)o;
  if (o == OUTH - 1) c -= 0.1;     // numpy: c[-1] -= 0.1
  return (long)floor(c / 1.5);
}

// ---------------- Kernel 1: weight generation (MLP + LN) + proj tables ----
__global__ void __launch_bounds__(256) nwgen_kernel(
    const float* __restrict__ w1, const float* __restrict__ b1,
    const float* __restrict__ w2, const float* __restrict__ b2,
    const float* __restrict__ lng, const float* __restrict__ lnb,
    float* __restrict__ ws)
{
  __shared__ float hbuf[9 * 256];
  __shared__ float wvbuf[KTOT];
  __shared__ float rs1[256], rs2[256];
  const int tid = threadIdx.x;

  // zero the padded B buffer (ws is poisoned, padding cols must be 0)
  for (int idx = tid; idx < B_SZ; idx += 256) ws[idx] = 0.0f;

  // projection tables (rows == cols here: same length, same scale)
  if (tid < OUTH) {
    int*   rsT = (int*)(ws + RS_OFF);
    float* vhT = ws + VH_OFF;
    long coord = coord_of(tid);
    int s = tid;
    while (s > 0 && coord_of(s - 1) == coord) --s;   // searchsorted
    long p = coord * 2 + (long)(tid - s);
    #pragma unroll
    for (int t = 0; t < 3; ++t) {
      long q = p + 2 * (t - 1);
      float val = (q >= 0 && q < (long)Hin * 2) ? 1.0f : 0.0f;
      long fd = (q < 0) ? (q - 1) / 2 : q / 2;        // floor(q/2)
      if (fd < 0) fd = 0;
      if (fd > Hin - 1) fd = Hin - 1;
      rsT[tid * 3 + t] = (int)fd;
      vhT[tid * 3 + t] = val;
    }
  }

  // hidden layer: h[pos][tid] = relu(posmat[pos] . w1[:,tid] + b1[tid])
  for (int pos = 0; pos < 9; ++pos) {
    int i = pos / 3, j = pos % 3;
    double oi = (double)i / 1.5; oi -= floor(oi);
    double oj = (double)j / 1.5; oj -= floor(oj);
    float p0 = (float)(1.0 / 1.5), p1 = (float)oi, p2 = (float)oj;
    float v = p0 * w1[tid] + p1 * w1[256 + tid] + p2 * w1[512 + tid] + b1[tid];
    hbuf[pos * 256 + tid] = v > 0.0f ? v : 0.0f;
  }
  __syncthreads();

  // output layer + layernorm over K per (pos, oc), scatter into B tiles
  for (int pos = 0; pos < 9; ++pos) {
    for (int oc = 0; oc < OUTC; ++oc) {
      for (int kk = tid; kk < KTOT; kk += 256) {
        const int jcol = oc * KTOT + kk;      // wflat index
        float acc = b2[jcol];
        const float* hp = hbuf + pos * 256;
        for (int r = 0; r < 256; ++r) acc += hp[r] * w2[r * 1728 + jcol];
        wvbuf[kk] = acc;
      }
      __syncthreads();

      float s = 0.0f, s2 = 0.0f;
      for (int kk = tid; kk < KTOT; kk += 256) {
        float v = wvbuf[kk]; s += v; s2 += v * v;
      }
      rs1[tid] = s; rs2[tid] = s2;
      __syncthreads();
      for (int st = 128; st > 0; st >>= 1) {
        if (tid < st) { rs1[tid] += rs1[tid + st]; rs2[tid] += rs2[tid + st]; }
        __syncthreads();
      }
      const float mu  = rs1[0] * (1.0f / KTOT);
      const float var = rs2[0] * (1.0f / KTOT) - mu * mu;
      const float inv = 1.0f / sqrtf(var + 1e-5f);

      for (int kk = tid; kk < KTOT; kk += 256) {
        float v = (wvbuf[kk] - mu) * inv * lng[kk] + lnb[kk];
        v *= (1.0f / INC);
        const int c  = kk / 9;
        const int r9 = kk - c * 9;            // t*3+u
        const int kp = r9 * INC + c;          // reordered K: (t,u) outer, c inner
        ws[(pos * KTOT + kp) * NPAD + oc] = v;
      }
      __syncthreads();
    }
  }
}

// ---------------- Kernel 2: gather + WMMA fp32 16x16x4 contraction --------
__global__ void __launch_bounds__(256) upsample_wmma_kernel(
    const float* __restrict__ x, const float* __restrict__ ws,
    float* __restrict__ out)
{
  __shared__ float ldsB[PHB];
  const int phase = blockIdx.x;          // 0..8
  const int i = phase / 3, j = phase % 3;

  for (int idx = threadIdx.x; idx < PHB; idx += 256)
    ldsB[idx] = ws[phase * PHB + idx];
  __syncthreads();

  const int*   rsT = (const int*)(ws + RS_OFF);
  const float* vhT = ws + VH_OFF;

  const int wave = threadIdx.x >> 5;
  const int lane = threadIdx.x & 31;
  const int half = lane >> 4;            // K half (A/B fragment layout)
  const int m    = lane & 15;            // M for A, N for B

  const int tile  = blockIdx.y * 8 + wave;   // 1152 tiles per phase
  const int pbase = tile * 16;
  const int pm  = pbase + m;                 // this lane's pixel (row M=m)
  const int n   = pm / 2304;                 // 48*48 pixels per image per phase
  const int rem = pm - n * 2304;
  const int ohp = rem / 48;
  const int owp = rem - ohp * 48;
  const int oh  = 3 * ohp + i;
  const int ow  = 3 * owp + j;

  const float* xn = x + (size_t)n * INC * HWs;
  int rofs[3]; float wh[3]; int cofs[3]; float wwt[3];
  #pragma unroll
  for (int t = 0; t < 3; ++t) { rofs[t] = rsT[oh * 3 + t]; wh[t]  = vhT[oh * 3 + t]; }
  #pragma unroll
  for (int u = 0; u < 3; ++u) { cofs[u] = rsT[ow * 3 + u]; wwt[u] = vhT[ow * 3 + u]; }

  v8f acc = {0.f, 0.f, 0.f, 0.f, 0.f, 0.f, 0.f, 0.f};

  #pragma unroll
  for (int t = 0; t < 3; ++t) {
    #pragma unroll
    for (int u = 0; u < 3; ++u) {
      const int   tu  = t * 3 + u;
      const float wtu = wh[t] * wwt[u];                 // 0/1 validity as weight
      const float* ap = xn + rofs[t] * Win + cofs[u] + (size_t)(2 * half) * HWs;
      const float* bp = ldsB + (tu * INC + 2 * half) * NPAD + m;
      #pragma unroll 4
      for (int s = 0; s < 16; ++s) {                    // K step of 4 channels
        v2f A, B;
        A.x = ap[(size_t)(4 * s)     * HWs] * wtu;      // A: K=2*half+0
        A.y = ap[(size_t)(4 * s + 1) * HWs] * wtu;      // A: K=2*half+1
        B.x = bp[(4 * s)     * NPAD];                    // B: same K, N=m
        B.y = bp[(4 * s + 1) * NPAD];
        acc = __builtin_amdgcn_wmma_f32_16x16x4_f32(
                  false, A, false, B, (short)0, acc, false, false);
      }
    }
  }

  // D layout: VGPR r, lanes 0-15 -> M=r, lanes 16-31 -> M=8+r; N = lane%16
  if (m < OUTC) {
    #pragma unroll
    for (int r = 0; r < 8; ++r) {
      const int pix = pbase + r + 8 * half;
      const int n2  = pix / 2304;
      const int r2  = pix - n2 * 2304;
      const int oh2 = 3 * (r2 / 48) + i;
      const int ow2 = 3 * (r2 - (r2 / 48) * 48) + j;
      out[(((size_t)n2 * OUTC + m) * OUTH + oh2) * OUTW + ow2] = acc[r];
    }
  }
}

extern "C" void kernel_launch(void* const* d_in, const int* in_sizes, int n_in,
                              void* d_out, int out_size, void* d_ws, size_t ws_size,
                              hipStream_t stream) {
  const float* x   = (const float*)d_in[0];
  const float* w1  = (const float*)d_in[1];
  const float* b1  = (const float*)d_in[2];
  const float* w2  = (const float*)d_in[3];
  const float* b2  = (const float*)d_in[4];
  const float* lng = (const float*)d_in[5];
  const float* lnb = (const float*)d_in[6];
  float* ws = (float*)d_ws;

  nwgen_kernel<<<1, 256, 0, stream>>>(w1, b1, w2, b2, lng, lnb, ws);
  upsample_wmma_kernel<<<dim3(9, 144, 1), 256, 0, stream>>>(x, ws, (float*)d_out);
}